// Attention_48765058679471
// MI455X (gfx1250) — compile-verified
//
#include <hip/hip_runtime.h>
#include <hip/hip_bf16.h>
#include <math.h>

typedef __attribute__((ext_vector_type(2))) float v2f;
typedef __attribute__((ext_vector_type(8))) float v8f;

#define B_ 32
#define S_ 2048
#define H_ 1024
#define KC 256            // K-chunk staged in LDS
#define ASTRIDE 260       // 256 + 4 pad -> conflict-free A-fragment loads
#define MT 32             // M rows (s positions) per block, 2 sub-tiles of 16
#define NCHUNK (H_ / KC)  // 4 K-chunks

// ---------------------------------------------------------------------------
// CDNA5 async global->LDS copy (ASYNCcnt-tracked; independent of LOADcnt so
// next-chunk staging does not serialize with the B-fragment load waits).
// GV mode: global_load_async_to_lds_b128 vdst(LDS byte offset), vaddr64, off
// ---------------------------------------------------------------------------
__device__ __forceinline__ void async_copy_b128(unsigned int lds_off,
                                                const float* gaddr) {
    asm volatile("global_load_async_to_lds_b128 %0, %1, off"
                 :: "v"(lds_off), "v"(gaddr)
                 : "memory");
}
__device__ __forceinline__ void wait_async0() {
    asm volatile("s_wait_asynccnt 0x0" ::: "memory");
}

// ---------------------------------------------------------------------------
// Kernel 0: repack Wk into WMMA B-fragment order. Fragment f = (h0/4)*64 + n0/16.
// Word lane*2+j  <->  B[k = 2*(lane>>4)+j][n = n0+(lane&15)], B[h][n] = Wk[n*H+h].
// ---------------------------------------------------------------------------
__global__ void pack_wk_kernel(const float* __restrict__ Wk,
                               float* __restrict__ WkP) {
    int lane = threadIdx.x & 31;
    int wave = threadIdx.x >> 5;
    int f    = blockIdx.x * 8 + wave;          // 16384 fragments total
    int h0   = (f >> 6) << 2;
    int n0   = (f & 63) << 4;
    int hi   = lane >> 4;
    int cn   = n0 + (lane & 15);
    float x = Wk[(size_t)cn * H_ + h0 + 2 * hi + 0];
    float y = Wk[(size_t)cn * H_ + h0 + 2 * hi + 1];
    float2 val = make_float2(x, y);
    *(float2*)(WkP + (size_t)f * 64 + lane * 2) = val;
}

// ---------------------------------------------------------------------------
// Kernel 1: q[b,k] = sum_h hidden[b,h] * Wq[k,h]
// ---------------------------------------------------------------------------
__global__ void query_kernel(const float* __restrict__ hidden,
                             const float* __restrict__ Wq,
                             float* __restrict__ q) {
    int k = blockIdx.x * blockDim.x + threadIdx.x;
    int b = blockIdx.y;
    const float4* hv = (const float4*)(hidden + (size_t)b * H_);
    const float4* wv = (const float4*)(Wq + (size_t)k * H_);
    float acc = 0.f;
    for (int i = 0; i < H_ / 4; ++i) {
        float4 h4 = hv[i];
        float4 w4 = wv[i];
        acc += h4.x * w4.x + h4.y * w4.y + h4.z * w4.z + h4.w * w4.w;
    }
    q[(size_t)b * H_ + k] = acc;
}

// ---------------------------------------------------------------------------
// Kernel 2: fused keys-GEMM + tanh + dot(v) -> scores[b,s]
// Block = (32 s-rows, batch b), 512 threads = 16 waves.
// Wave w owns columns [w*64, w*64+64) as 4 n-tiles; each B fragment feeds
// both 16-row M sub-tiles (2 WMMAs per global b64 load).
// enc chunks double-buffered in LDS via global_load_async_to_lds_b128.
// ---------------------------------------------------------------------------
__global__ void __launch_bounds__(512, 1)
attn_scores_kernel(const float* __restrict__ enc,
                   const float* __restrict__ WkP,
                   const float* __restrict__ q,
                   const float* __restrict__ v,
                   float* __restrict__ scores) {
    __shared__ float sA[2][MT * ASTRIDE];
    __shared__ float sAttn[MT];

    const int tid  = threadIdx.x;
    const int lane = tid & 31;
    const int wave = tid >> 5;
    const int b    = blockIdx.y;
    const int s0   = blockIdx.x * MT;
    const int m    = lane & 15;
    const int hi   = lane >> 4;

    if (tid < MT) sAttn[tid] = 0.f;

    v8f acc0[4] = {};     // rows s0 .. s0+15
    v8f acc1[4] = {};     // rows s0+16 .. s0+31

    const float* encRow = enc + ((size_t)b * S_ + s0) * H_;

    // per-thread staging: 4 async b128 copies = 32x256 floats per chunk
    const int srow = tid >> 7;          // base row group: 4 iters advance by 8
    const int sc4  = tid & 127;         // unused name guard
    (void)srow; (void)sc4;

    auto stage = [&](int buf, int kc) {
        #pragma unroll
        for (int i = 0; i < 4; ++i) {
            int idx = tid + i * 512;    // 0..2047 float4 slots
            int row = idx >> 6;
            int c4  = idx & 63;
            const float* gp = encRow + (size_t)row * H_ + kc * KC + c4 * 4;
            unsigned int loff =
                (unsigned int)(uintptr_t)&sA[buf][row * ASTRIDE + c4 * 4];
            async_copy_b128(loff, gp);
        }
    };

    stage(0, 0);
    wait_async0();
    __syncthreads();

    for (int kc = 0; kc < NCHUNK; ++kc) {
        const int cur = kc & 1;
        if (kc + 1 < NCHUNK)
            stage(cur ^ 1, kc + 1);     // overlaps with WMMA loop below

        const float* curA = sA[cur];
        for (int ks = 0; ks < KC / 4; ++ks) {
            // A fragments: 16x4 fp32; lanes<16 K=+0/+1, lanes>=16 K=+2/+3
            v2f a0 = *(const v2f*)(&curA[m * ASTRIDE + ks * 4 + 2 * hi]);
            v2f a1 = *(const v2f*)(&curA[(m + 16) * ASTRIDE + ks * 4 + 2 * hi]);
            const float* bp = WkP +
                ((size_t)((kc * 64 + ks) * 64 + wave * 4) * 64) + lane * 2;
            // batch the 4 B-fragment loads so they pipeline ahead of the WMMAs
            v2f bb[4];
            #pragma unroll
            for (int t = 0; t < 4; ++t)
                bb[t] = *(const v2f*)(bp + t * 64);
            #pragma unroll
            for (int t = 0; t < 4; ++t) {
                acc0[t] = __builtin_amdgcn_wmma_f32_16x16x4_f32(
                    false, a0, false, bb[t], (short)0, acc0[t], false, false);
                acc1[t] = __builtin_amdgcn_wmma_f32_16x16x4_f32(
                    false, a1, false, bb[t], (short)0, acc1[t], false, false);
            }
        }

        if (kc + 1 < NCHUNK) {
            wait_async0();              // own async copies for kc+1 done
            __syncthreads();            // visible to all waves; cur buf free
        }
    }

    // Epilogue: attn[s] += tanh(q[col] + keys[s,col]) * v[col]
    // C layout: (vgpr r, lane) = keys[row = r + 8*hi (+16 for acc1)][col = lane&15]
    float part0[8], part1[8];
    #pragma unroll
    for (int r = 0; r < 8; ++r) { part0[r] = 0.f; part1[r] = 0.f; }

    #pragma unroll
    for (int t = 0; t < 4; ++t) {
        int col = wave * 64 + t * 16 + m;
        float qc = q[(size_t)b * H_ + col];
        float vc = v[col];
        #pragma unroll
        for (int r = 0; r < 8; ++r) {
            part0[r] += tanhf(qc + acc0[t][r]) * vc;
            part1[r] += tanhf(qc + acc1[t][r]) * vc;
        }
    }

    // reduce over the 16 lanes of each half (columns of this wave's band)
    #pragma unroll
    for (int r = 0; r < 8; ++r) {
        float x = part0[r];
        x += __shfl_xor(x, 1, 32);
        x += __shfl_xor(x, 2, 32);
        x += __shfl_xor(x, 4, 32);
        x += __shfl_xor(x, 8, 32);
        part0[r] = x;
        float y = part1[r];
        y += __shfl_xor(y, 1, 32);
        y += __shfl_xor(y, 2, 32);
        y += __shfl_xor(y, 4, 32);
        y += __shfl_xor(y, 8, 32);
        part1[r] = y;
    }

    if (m == 0) {
        #pragma unroll
        for (int r = 0; r < 8; ++r) {
            atomicAdd(&sAttn[hi * 8 + r], part0[r]);        // ds_add_f32
            atomicAdd(&sAttn[16 + hi * 8 + r], part1[r]);
        }
    }
    __syncthreads();
    if (tid < MT) scores[(size_t)b * S_ + s0 + tid] = sAttn[tid];
}

// ---------------------------------------------------------------------------
// Kernel 3: masked, stable softmax over S per batch row
// ---------------------------------------------------------------------------
__global__ void softmax_kernel(const float* __restrict__ scores,
                               const long long* __restrict__ lengths,
                               float* __restrict__ out) {
    __shared__ float red[256];
    int b = blockIdx.x;
    int tid = threadIdx.x;
    int len = (int)lengths[b];

    float vals[8];
    float mx = -INFINITY;
    #pragma unroll
    for (int i = 0; i < 8; ++i) {
        int s = tid + i * 256;
        float sc = scores[(size_t)b * S_ + s];
        vals[i] = (s < len) ? sc : -INFINITY;
        mx = fmaxf(mx, vals[i]);
    }
    red[tid] = mx;
    __syncthreads();
    for (int o = 128; o > 0; o >>= 1) {
        if (tid < o) red[tid] = fmaxf(red[tid], red[tid + o]);
        __syncthreads();
    }
    mx = red[0];
    __syncthreads();

    float e[8];
    float sum = 0.f;
    #pragma unroll
    for (int i = 0; i < 8; ++i) {
        e[i] = (vals[i] == -INFINITY) ? 0.f : __expf(vals[i] - mx);
        sum += e[i];
    }
    red[tid] = sum;
    __syncthreads();
    for (int o = 128; o > 0; o >>= 1) {
        if (tid < o) red[tid] += red[tid + o];
        __syncthreads();
    }
    float inv = 1.0f / red[0];
    #pragma unroll
    for (int i = 0; i < 8; ++i)
        out[(size_t)b * S_ + tid + i * 256] = e[i] * inv;
}

// ---------------------------------------------------------------------------
extern "C" void kernel_launch(void* const* d_in, const int* in_sizes, int n_in,
                              void* d_out, int out_size, void* d_ws, size_t ws_size,
                              hipStream_t stream) {
    const float*     hidden  = (const float*)d_in[0];
    const float*     enc     = (const float*)d_in[1];
    const long long* lengths = (const long long*)d_in[2];
    const float*     Wq      = (const float*)d_in[3];
    const float*     Wk      = (const float*)d_in[4];
    const float*     v       = (const float*)d_in[5];
    float* out = (float*)d_out;

    float* WkP    = (float*)d_ws;                  // H*H floats (4 MB)
    float* q      = WkP + (size_t)H_ * H_;         // B*H floats
    float* scores = q + (size_t)B_ * H_;           // B*S floats

    pack_wk_kernel<<<2048, 256, 0, stream>>>(Wk, WkP);
    query_kernel<<<dim3(H_ / 256, B_), 256, 0, stream>>>(hidden, Wq, q);
    attn_scores_kernel<<<dim3(S_ / MT, B_), 512, 0, stream>>>(enc, WkP, q, v, scores);
    softmax_kernel<<<B_, 256, 0, stream>>>(scores, lengths, out);
}